// E3PhisNet_29609504538691
// MI455X (gfx1250) — compile-verified
//
#include <hip/hip_runtime.h>
#include <math.h>

// ---------------------------------------------------------------------------
// E3PhisNet matrix-block pipeline for MI455X (gfx1250, wave32).
// Channel-mixing GEMMs (K=N=128) run on V_WMMA_F32_16X16X4_F32.
// CG tensor-product ops are VALU with the CG table staged in LDS (62.5KB).
// ---------------------------------------------------------------------------

#define N_NODES 512
#define N_EDGES 16384
#define CCH     128
#define FDIST   32
#define PPM     15           // len(paths(2,2,2))
#define CG_TOTAL 15625       // sum over l1,l2,l3<=4 of (2l1+1)(2l2+1)(2l3+1)

typedef float v2f __attribute__((ext_vector_type(2)));
typedef float v8f __attribute__((ext_vector_type(8)));

__host__ __device__ inline int cg_off(int l1, int l2, int l3) {
  // layout: l1 outer, l2, l3; element (i,j,m) row-major
  return 625 * l1 * l1 + 25 * (2 * l1 + 1) * l2 * l2 +
         (2 * l1 + 1) * (2 * l2 + 1) * l3 * l3;
}

__device__ inline float silu_f(float x) { return x / (1.f + __expf(-x)); }

// ======================= CG coefficient init ===============================
__device__ double dfact(int n) { double r = 1.0; for (int i = 2; i <= n; ++i) r *= (double)i; return r; }

__device__ double su2cg(int j1, int m1, int j2, int m2, int j3, int m3) {
  if (m3 != m1 + m2) return 0.0;
  int vmin = max(max(-j1 + j2 + m3, -j1 + m1), 0);
  int vmax = min(min(j2 + j3 + m1, j3 - j1 + j2), j3 + m3);
  double C = sqrt((double)(2 * j3 + 1) * dfact(j3 + j1 - j2) * dfact(j3 - j1 + j2) *
                  dfact(j1 + j2 - j3) * dfact(j3 + m3) * dfact(j3 - m3) /
                  (dfact(j1 + j2 + j3 + 1) * dfact(j1 - m1) * dfact(j1 + m1) *
                   dfact(j2 - m2) * dfact(j2 + m2)));
  double S = 0.0;
  for (int v = vmin; v <= vmax; ++v) {
    double sgn = ((v + j2 + m2) & 1) ? -1.0 : 1.0;
    S += sgn / dfact(v) * dfact(j2 + j3 + m1 - v) * dfact(j1 - m1 + v) /
         (dfact(j3 - j1 + j2 - v) * dfact(j3 + m3 - v) * dfact(v + j1 - j2 - m3));
  }
  return C * S;
}

struct cplx { double re, im; };
__device__ inline cplx cmul(cplx a, cplx b) { return {a.re * b.re - a.im * b.im, a.re * b.im + a.im * b.re}; }

// row r of q_real_to_complex(l): at most 2 nonzeros; includes (-1j)^l factor
__device__ int qrow(int l, int r, int* cols, cplx* vals) {
  const double is2 = 0.70710678118654752440;
  int m = r - l, n;
  cplx v0 = {0, 0}, v1 = {0, 0};
  if (m < 0)      { cols[0] = l - m; v0 = {is2, 0}; cols[1] = l + m; v1 = {0, -is2}; n = 2; }
  else if (m == 0){ cols[0] = l;     v0 = {1.0, 0}; n = 1; }
  else            { double s = (m & 1) ? -1.0 : 1.0;
                    cols[0] = l + m; v0 = {s * is2, 0}; cols[1] = l - m; v1 = {0, s * is2}; n = 2; }
  cplx ph;
  switch (l & 3) { case 0: ph = {1, 0}; break; case 1: ph = {0, -1}; break;
                   case 2: ph = {-1, 0}; break; default: ph = {0, 1}; }
  vals[0] = cmul(v0, ph);
  if (n > 1) vals[1] = cmul(v1, ph);
  return n;
}

__global__ __launch_bounds__(256) void kcginit(float* __restrict__ cg) {
  int t = blockIdx.x * 256 + threadIdx.x;
  if (t >= CG_TOTAL) return;
  int rem = t, l1 = 0, l2 = 0, l3 = 0;
  for (;;) { int sz = 625 * (2 * l1 + 1); if (rem < sz) break; rem -= sz; ++l1; }
  for (;;) { int sz = 25 * (2 * l1 + 1) * (2 * l2 + 1); if (rem < sz) break; rem -= sz; ++l2; }
  for (;;) { int sz = (2 * l1 + 1) * (2 * l2 + 1) * (2 * l3 + 1); if (rem < sz) break; rem -= sz; ++l3; }
  int n2 = 2 * l2 + 1, n3 = 2 * l3 + 1;
  int i = rem / (n2 * n3), k = (rem / n3) % n2, m = rem % n3;
  float out = 0.f;
  int dlo = (l1 > l2) ? l1 - l2 : l2 - l1;
  if (l3 >= dlo && l3 <= l1 + l2) {
    int c1[2], c2[2], c3[2]; cplx v1[2], v2[2], v3[2];
    int na = qrow(l1, i, c1, v1), nb = qrow(l2, k, c2, v2), nc = qrow(l3, m, c3, v3);
    cplx s = {0, 0};
    for (int a = 0; a < na; ++a)
      for (int b = 0; b < nb; ++b)
        for (int c = 0; c < nc; ++c) {
          double cc = su2cg(l1, c1[a] - l1, l2, c2[b] - l2, l3, c3[c] - l3);
          if (cc != 0.0) {
            cplx v3c = {v3[c].re, -v3[c].im};                  // conj(Q3)
            cplx p = cmul(cmul(v1[a], v2[b]), v3c);
            s.re += p.re * cc; s.im += p.im * cc;
          }
        }
    out = (float)s.re;
  }
  cg[t] = out;
}

// ======================= channel_norm (+ optional silu on l=0) =============
__global__ __launch_bounds__(256) void knorm(const float* __restrict__ F, float* __restrict__ O,
                                             const float* __restrict__ g, int S, int L, int doact) {
  int b = blockIdx.x, tid = threadIdx.x;
  __shared__ float red[256];
  __shared__ float nl[8];
  const float* f = F + (size_t)b * CCH * S;
  for (int l = 0; l <= L; ++l) {
    int n3 = 2 * l + 1, cnt = CCH * n3;
    float s = 0.f;
    for (int idx = tid; idx < cnt; idx += 256) {
      int c = idx / n3, m = idx - c * n3;
      float v = f[(size_t)c * S + l * l + m];
      s += v * v;
    }
    red[tid] = s; __syncthreads();
    for (int st = 128; st > 0; st >>= 1) { if (tid < st) red[tid] += red[tid + st]; __syncthreads(); }
    if (tid == 0) nl[l] = sqrtf(red[0] / (float)cnt + 1e-6f);
    __syncthreads();
  }
  for (int idx = tid; idx < CCH * S; idx += 256) {
    int c = idx / S, i = idx - c * S;
    int l = (i >= 1) + (i >= 4) + (i >= 9) + (i >= 16);
    float v = f[(size_t)c * S + i] / nl[l] * g[c * (L + 1) + l];
    if (doact && i == 0) v = silu_f(v);
    O[(size_t)b * CCH * S + idx] = v;
  }
}

// ======================= selfmix (TP(x,x) path-weighted) ===================
__global__ __launch_bounds__(256) void kselfmix(const float* __restrict__ Z, float* __restrict__ O,
                                                const float* __restrict__ wv, const float* __restrict__ kmat,
                                                const float* __restrict__ bsc, const float* __restrict__ cg,
                                                int B, int Lx, int Lout) {
  __shared__ float scg[CG_TOTAL];
  __shared__ float sw[80];
  for (int t = threadIdx.x; t < CG_TOTAL; t += 256) scg[t] = cg[t];
  int P = 0;
  for (int a = 0; a <= Lx; ++a)
    for (int b2 = 0; b2 <= Lx; ++b2)
      for (int c3 = 0; c3 <= Lout; ++c3) {
        int dlo = (a > b2) ? a - b2 : b2 - a;
        if (c3 >= dlo && c3 <= a + b2) ++P;
      }
  for (int t = threadIdx.x; t < P; t += 256) sw[t] = wv[t];
  __syncthreads();
  int gid = blockIdx.x * 256 + threadIdx.x;
  if (gid >= B * CCH) return;
  int b = gid >> 7, c = gid & 127;
  int Sz = (Lx + 1) * (Lx + 1), So = (Lout + 1) * (Lout + 1);
  float z[25];
  for (int i = 0; i < Sz; ++i) z[i] = Z[((size_t)b * CCH + c) * Sz + i];
  int kc = Lx + 1;  // min(Lx,Lout)+1 ; all our uses have Lout>=Lx
  for (int l3 = 0; l3 <= Lout; ++l3) {
    int n3 = 2 * l3 + 1;
    float acc[9];
    for (int m = 0; m < n3; ++m) acc[m] = 0.f;
    int p = 0;
    for (int l1 = 0; l1 <= Lx; ++l1)
      for (int l2 = 0; l2 <= Lx; ++l2)
        for (int t3 = 0; t3 <= Lout; ++t3) {
          int dlo = (l1 > l2) ? l1 - l2 : l2 - l1;
          if (t3 < dlo || t3 > l1 + l2) continue;
          if (t3 == l3) {
            float wp = sw[p];
            const float* gp = scg + cg_off(l1, l2, l3);
            int n1 = 2 * l1 + 1, n2 = 2 * l2 + 1;
            for (int i = 0; i < n1; ++i) {
              float zi = z[l1 * l1 + i];
              for (int j = 0; j < n2; ++j) {
                float f = wp * zi * z[l2 * l2 + j];
                const float* gr = gp + (i * n2 + j) * n3;
                for (int m = 0; m < n3; ++m) acc[m] += gr[m] * f;
              }
            }
          }
          ++p;
        }
    if (l3 <= Lx) {
      float kk = kmat[c * kc + l3];
      for (int m = 0; m < n3; ++m) acc[m] += kk * z[l3 * l3 + m];
    }
    if (l3 == 0) acc[0] += bsc[0];
    for (int m = 0; m < n3; ++m) O[((size_t)b * CCH + c) * So + l3 * l3 + m] = acc[m];
  }
}

// ======================= channel_mix GEMM via WMMA f32 16x16x4 =============
// out[b,d,i] = sum_c z[b,c,i] * W[c,d]  (+bias at l=0, opt silu, opt skip)
// flags: 1=bias(l==0 launch), 2=silu, 4=skip-add
__global__ __launch_bounds__(128) void kchanmix(const float* __restrict__ Z, float* __restrict__ O,
                                                const float* __restrict__ W, const float* __restrict__ bias,
                                                const float* __restrict__ skip,
                                                int Sz, int So, int Sskip, int l, int flags) {
  const int comps = 2 * l + 1, base = l * l;
  const int wvi = threadIdx.x >> 5, lane = threadIdx.x & 31;
  const int lm = lane & 15, lh = lane >> 4;
  const int r0 = blockIdx.x * 16;
  const int d0 = (blockIdx.y * 4 + wvi) * 16;
  const int rA = r0 + lm;
  const int bA = rA / comps;
  const int iA = base + (rA - bA * comps);
  const float* zr = Z + ((size_t)bA * CCH) * Sz + iA;
  v8f acc = {};
  for (int kk = 0; kk < CCH; kk += 4) {
    const int ka = kk + lh * 2;
    v2f a, bfrag;
    a.x = zr[(size_t)(ka + 0) * Sz];
    a.y = zr[(size_t)(ka + 1) * Sz];
    bfrag.x = W[(size_t)(ka + 0) * CCH + d0 + lm];
    bfrag.y = W[(size_t)(ka + 1) * CCH + d0 + lm];
    acc = __builtin_amdgcn_wmma_f32_16x16x4_f32(false, a, false, bfrag, (short)0, acc, false, false);
  }
  const int d = d0 + lm;
  const float bv = (flags & 1) ? bias[d] : 0.f;
  for (int v = 0; v < 8; ++v) {
    const int r = r0 + v + lh * 8;
    const int bb = r / comps;
    const int i = base + (r - bb * comps);
    float x = acc[v] + bv;
    if (flags & 2) x = silu_f(x);
    if (flags & 4) x += skip[((size_t)bb * CCH + d) * Sskip + i];
    O[((size_t)bb * CCH + d) * So + i] = x;
  }
}

// ======================= dist @ Wd + bd (E x 32 x 1920) via WMMA ===========
__global__ __launch_bounds__(128) void kgemm_dist(const float* __restrict__ dist, const float* __restrict__ Wd,
                                                  const float* __restrict__ bd, float* __restrict__ out) {
  const int wvi = threadIdx.x >> 5, lane = threadIdx.x & 31;
  const int lm = lane & 15, lh = lane >> 4;
  const int r0 = blockIdx.x * 16;
  const int n0 = (blockIdx.y * 4 + wvi) * 16;
  v8f acc = {};
  for (int kk = 0; kk < FDIST; kk += 4) {
    const int ka = kk + lh * 2;
    v2f a, bfrag;
    a.x = dist[(size_t)(r0 + lm) * FDIST + ka + 0];
    a.y = dist[(size_t)(r0 + lm) * FDIST + ka + 1];
    bfrag.x = Wd[(size_t)(ka + 0) * (CCH * PPM) + n0 + lm];
    bfrag.y = Wd[(size_t)(ka + 1) * (CCH * PPM) + n0 + lm];
    acc = __builtin_amdgcn_wmma_f32_16x16x4_f32(false, a, false, bfrag, (short)0, acc, false, false);
  }
  const int n = n0 + lm;
  const float bv = bd[n];
  for (int v = 0; v < 8; ++v) {
    const int r = r0 + v + lh * 8;
    out[(size_t)r * (CCH * PPM) + n] = acc[v] + bv;
  }
}

// ======================= pairmix (edge TP with per-edge weights) ===========
__global__ __launch_bounds__(256) void kpairmix(const float* __restrict__ cbuf, const float* __restrict__ nbuf,
                                                const int* __restrict__ ei, const float* __restrict__ wbuf,
                                                const float* __restrict__ cg, float* __restrict__ out) {
  __shared__ float scg[CG_TOTAL];
  for (int t = threadIdx.x; t < CG_TOTAL; t += 256) scg[t] = cg[t];
  __syncthreads();
  int gid = blockIdx.x * 256 + threadIdx.x;
  if (gid >= N_EDGES * CCH) return;
  int e = gid >> 7, c = gid & 127;
  int src = ei[2 * e + 0], dst = ei[2 * e + 1];
  float x[9], y[9], w[PPM], acc[9];
  const float* xr = cbuf + ((size_t)dst * CCH + c) * 9;
  const float* yr = nbuf + ((size_t)src * CCH + c) * 9;
  for (int i = 0; i < 9; ++i) { x[i] = xr[i]; y[i] = yr[i]; acc[i] = 0.f; }
  const float* wr = wbuf + (size_t)e * (CCH * PPM) + c * PPM;
  for (int p = 0; p < PPM; ++p) w[p] = wr[p];
  int p = 0;
  for (int l1 = 0; l1 <= 2; ++l1)
    for (int l2 = 0; l2 <= 2; ++l2)
      for (int l3 = 0; l3 <= 2; ++l3) {
        int dlo = (l1 > l2) ? l1 - l2 : l2 - l1;
        if (l3 < dlo || l3 > l1 + l2) continue;
        const float* gp = scg + cg_off(l1, l2, l3);
        float wp = w[p];
        int n1 = 2 * l1 + 1, n2 = 2 * l2 + 1, n3 = 2 * l3 + 1;
        for (int i = 0; i < n1; ++i)
          for (int j = 0; j < n2; ++j) {
            float f = wp * x[l1 * l1 + i] * y[l2 * l2 + j];
            const float* gr = gp + (i * n2 + j) * n3;
            for (int m = 0; m < n3; ++m) acc[l3 * l3 + m] += gr[m] * f;
          }
        ++p;
      }
  float* o = out + ((size_t)e * CCH + c) * 9;
  for (int i = 0; i < 9; ++i) o[i] = acc[i];
}

// ======================= lin_mul (channels -> per-l multiplicity) ==========
__device__ const int D_MULS[5] = {6, 8, 8, 3, 1};
__device__ const int D_LOFF[6] = {0, 6, 30, 70, 91, 100};
struct WPtrs { const float* W[5]; };

__global__ __launch_bounds__(256) void klinmul(const float* __restrict__ Z, float* __restrict__ O,
                                               WPtrs w, const float* __restrict__ b0, int B) {
  int gid = blockIdx.x * 256 + threadIdx.x;
  if (gid >= B * 100) return;
  int b = gid / 100, s = gid - b * 100;
  int l = 0;
  while (l < 4 && s >= D_LOFF[l + 1]) ++l;
  int rem = s - D_LOFF[l], n3 = 2 * l + 1;
  int d = rem / n3, m = rem - d * n3;
  const float* zr = Z + (size_t)b * CCH * 25 + l * l + m;
  const float* wl = w.W[l];
  float acc = 0.f;
  for (int c = 0; c < CCH; ++c) acc += zr[(size_t)c * 25] * wl[c * D_MULS[l] + d];
  if (l == 0) acc += b0[d];
  O[(size_t)b * 100 + s] = acc;
}

// ======================= block mapping (irreps -> 10x10 blocks) ============
__device__ const int D_BL[4]   = {0, 0, 1, 2};
__device__ const int D_SLOT[4] = {0, 1, 2, 5};
__device__ const int D_BIDX[16][5] = {
  { 0,-1,-1,-1,-1}, { 1,-1,-1,-1,-1}, {-1, 0,-1,-1,-1}, {-1,-1, 0,-1,-1},
  { 2,-1,-1,-1,-1}, { 3,-1,-1,-1,-1}, {-1, 1,-1,-1,-1}, {-1,-1, 1,-1,-1},
  {-1, 2,-1,-1,-1}, {-1, 3,-1,-1,-1}, { 4, 4, 2,-1,-1}, {-1, 5, 3, 0,-1},
  {-1,-1, 4,-1,-1}, {-1,-1, 5,-1,-1}, {-1, 6, 6, 1,-1}, { 5, 7, 7, 2, 0}};

__global__ __launch_bounds__(256) void kblockmap(const float* __restrict__ feat, const float* __restrict__ cg,
                                                 float* __restrict__ blocks, int B) {
  int gid = blockIdx.x * 256 + threadIdx.x;
  if (gid >= B * 16) return;
  int b = gid >> 4, pr = gid & 15;
  int a = pr >> 2, bb = pr & 3;
  int lr = D_BL[a], lc = D_BL[bb];
  int nr = 2 * lr + 1, nc = 2 * lc + 1;
  float blk[25];
  for (int t = 0; t < nr * nc; ++t) blk[t] = 0.f;
  int dlo = (lr > lc) ? lr - lc : lc - lr;
  for (int l3 = dlo; l3 <= lr + lc; ++l3) {
    int idx = D_BIDX[pr][l3];
    int n3 = 2 * l3 + 1;
    const float* x = feat + (size_t)b * 100 + D_LOFF[l3] + idx * n3;
    const float* g = cg + cg_off(lr, lc, l3);
    for (int i = 0; i < nr; ++i)
      for (int j = 0; j < nc; ++j) {
        float s = 0.f;
        const float* gr = g + (i * nc + j) * n3;
        for (int m = 0; m < n3; ++m) s += gr[m] * x[m];
        blk[i * nc + j] += s;
      }
  }
  for (int i = 0; i < nr; ++i)
    for (int j = 0; j < nc; ++j)
      blocks[(size_t)b * 100 + (D_SLOT[a] + i) * 10 + (D_SLOT[bb] + j)] = blk[i * nc + j];
}

// ======================= assembly ==========================================
__global__ __launch_bounds__(256) void kscatter_diag(const float* __restrict__ dblk, float* __restrict__ Out) {
  int gid = blockIdx.x * 256 + threadIdx.x;
  if (gid >= N_NODES * 100) return;
  int i = gid / 100, pq = gid - i * 100;
  Out[((size_t)i * N_NODES + i) * 100 + pq] = dblk[gid];
}

__global__ __launch_bounds__(128) void kscatter_off(const float* __restrict__ oblk, const int* __restrict__ ei,
                                                    float* __restrict__ Out) {
  int e = blockIdx.x;
  int src = ei[2 * e + 0], dst = ei[2 * e + 1];
  __shared__ int dup;
  if (threadIdx.x == 0) dup = 0;
  __syncthreads();
  for (int e2 = e + 1 + threadIdx.x; e2 < N_EDGES; e2 += 128) {  // last-wins .set semantics
    if (ei[2 * e2] == src && ei[2 * e2 + 1] == dst) { dup = 1; break; }
  }
  __syncthreads();
  if (dup) return;
  for (int t = threadIdx.x; t < 100; t += 128)
    Out[((size_t)dst * N_NODES + src) * 100 + t] = oblk[(size_t)e * 100 + t];
}

__global__ __launch_bounds__(128) void ksym(float* __restrict__ Out) {
  int i = blockIdx.x, j = blockIdx.y;
  if (j < i) return;
  int t = threadIdx.x;
  if (t >= 100) return;
  int p = t / 10, q = t - p * 10;
  if (i == j) {
    if (q < p) return;
    size_t base = ((size_t)i * N_NODES + i) * 100;
    float a = Out[base + p * 10 + q], b = Out[base + q * 10 + p];
    float v = 0.5f * (a + b);
    Out[base + p * 10 + q] = v; Out[base + q * 10 + p] = v;
    return;
  }
  size_t bij = ((size_t)i * N_NODES + j) * 100;
  size_t bji = ((size_t)j * N_NODES + i) * 100;
  float a = Out[bij + p * 10 + q], b = Out[bji + q * 10 + p];
  float v = 0.5f * (a + b);
  Out[bij + p * 10 + q] = v; Out[bji + q * 10 + p] = v;
}

// ======================= host orchestration ================================
// Param leaf order: top-level dict in insertion order, nested params flattened
// with sorted keys (jax tree flatten convention):
// 0 nodes, 1 edge_ind, 2 dist
// 3 g_diag, 4 g_off
// 5..9 lin_diag.W[0..4], 10 lin_diag.b ; 11..15 lin_offdiag.W, 16 b
// 17 pairmix.Wd, 18 pairmix.bd
// res_center @19 (15 leaves: g, lin1{cm.W0..2, cm.b, sm.b, sm.k, sm.w}, lin2{...})
// res_diag   @34 (19 leaves: g, lin1{cm.W0..4, cm.b, sm.b, sm.k, sm.w}, lin2{...})
// res_neighbor @53 (15), res_offdiag @68 (19)
enum { RC = 19, RD = 34, RN = 53, RO = 68 };

extern "C" void kernel_launch(void* const* d_in, const int* in_sizes, int n_in,
                              void* d_out, int out_size, void* d_ws, size_t ws_size,
                              hipStream_t stream) {
  (void)in_sizes; (void)n_in; (void)ws_size;
  auto F = [&](int i) { return (const float*)d_in[i]; };
  const float* nodes = F(0);
  const int* ei = (const int*)d_in[1];
  const float* dist = F(2);
  float* out = (float*)d_out;

  float* ws = (float*)d_ws;
  size_t off = 0;
  auto alloc = [&](size_t n) { float* p = ws + off; off += n; return p; };
  float* CG = alloc(CG_TOTAL);
  float* NZ = alloc((size_t)N_NODES * CCH * 9);
  float* NS = alloc((size_t)N_NODES * CCH * 25);
  float* NT = alloc((size_t)N_NODES * CCH * 25);
  float* CB = alloc((size_t)N_NODES * CCH * 9);
  float* NB = alloc((size_t)N_NODES * CCH * 9);
  float* DI = alloc((size_t)N_NODES * CCH * 25);
  float* DF = alloc((size_t)N_NODES * 100);
  float* DB = alloc((size_t)N_NODES * 100);
  float* E9 = alloc((size_t)N_EDGES * CCH * 9);
  float* EZ = alloc((size_t)N_EDGES * CCH * 9);
  float* EA = alloc((size_t)N_EDGES * CCH * 25);
  float* EB = alloc((size_t)N_EDGES * CCH * 25);
  float* OF = alloc((size_t)N_EDGES * 100);
  float* OB = alloc((size_t)N_EDGES * 100);
  float* WB = EA;  // dist-weight buffer (E x 1920) aliases EA (dead by then)

  kcginit<<<(CG_TOTAL + 255) / 256, 256, 0, stream>>>(CG);

  auto chanmix = [&](const float* Z, float* O, const float* W, const float* bias,
                     const float* skip, int B, int Sz, int So, int Sskip, int l, int flags) {
    dim3 g((B * (2 * l + 1)) / 16, 2);
    kchanmix<<<g, 128, 0, stream>>>(Z, O, W, bias, skip, Sz, So, Sskip, l, flags);
  };

  // --- residual L2->L2 on nodes (res_center / res_neighbor) ---
  auto run_res22 = [&](const float* fin, int base, float* outb) {
    knorm<<<N_NODES, 256, 0, stream>>>(fin, NZ, F(base + 0), 9, 2, 1);
    kselfmix<<<(N_NODES * CCH) / 256, 256, 0, stream>>>(NZ, NS, F(base + 7), F(base + 6), F(base + 5), CG, N_NODES, 2, 2);
    for (int l = 0; l <= 2; ++l)
      chanmix(NS, NT, F(base + 1 + l), F(base + 4), nullptr, N_NODES, 9, 9, 0, l, l == 0 ? 3 : 0);
    kselfmix<<<(N_NODES * CCH) / 256, 256, 0, stream>>>(NT, NS, F(base + 14), F(base + 13), F(base + 12), CG, N_NODES, 2, 2);
    for (int l = 0; l <= 2; ++l)
      chanmix(NS, outb, F(base + 8 + l), F(base + 11), fin, N_NODES, 9, 9, 9, l, (l == 0 ? 1 : 0) | 4);
  };
  run_res22(nodes, RC, CB);
  run_res22(nodes, RN, NB);

  // --- res_diag: L2 -> L4 on center features ---
  knorm<<<N_NODES, 256, 0, stream>>>(CB, NZ, F(RD + 0), 9, 2, 1);
  kselfmix<<<(N_NODES * CCH) / 256, 256, 0, stream>>>(NZ, NS, F(RD + 9), F(RD + 8), F(RD + 7), CG, N_NODES, 2, 4);
  for (int l = 0; l <= 4; ++l)
    chanmix(NS, NT, F(RD + 1 + l), F(RD + 6), nullptr, N_NODES, 25, 25, 0, l, l == 0 ? 3 : 0);
  kselfmix<<<(N_NODES * CCH) / 256, 256, 0, stream>>>(NT, NS, F(RD + 18), F(RD + 17), F(RD + 16), CG, N_NODES, 4, 4);
  for (int l = 0; l <= 4; ++l)
    chanmix(NS, DI, F(RD + 10 + l), F(RD + 15), CB, N_NODES, 25, 25, 9, l, (l == 0 ? 1 : 0) | (l <= 2 ? 4 : 0));
  knorm<<<N_NODES, 256, 0, stream>>>(DI, NT, F(3), 25, 4, 0);
  { WPtrs w = {{F(5), F(6), F(7), F(8), F(9)}};
    klinmul<<<(N_NODES * 100 + 255) / 256, 256, 0, stream>>>(NT, DF, w, F(10), N_NODES); }
  kblockmap<<<(N_NODES * 16 + 255) / 256, 256, 0, stream>>>(DF, CG, DB, N_NODES);

  // --- edge pipeline ---
  kgemm_dist<<<dim3(N_EDGES / 16, (CCH * PPM) / 64), 128, 0, stream>>>(dist, F(17), F(18), WB);
  kpairmix<<<(N_EDGES * CCH) / 256, 256, 0, stream>>>(CB, NB, ei, WB, CG, E9);
  knorm<<<N_EDGES, 256, 0, stream>>>(E9, EZ, F(RO + 0), 9, 2, 1);
  kselfmix<<<(N_EDGES * CCH) / 256, 256, 0, stream>>>(EZ, EA, F(RO + 9), F(RO + 8), F(RO + 7), CG, N_EDGES, 2, 4);
  for (int l = 0; l <= 4; ++l)
    chanmix(EA, EB, F(RO + 1 + l), F(RO + 6), nullptr, N_EDGES, 25, 25, 0, l, l == 0 ? 3 : 0);
  kselfmix<<<(N_EDGES * CCH) / 256, 256, 0, stream>>>(EB, EA, F(RO + 18), F(RO + 17), F(RO + 16), CG, N_EDGES, 4, 4);
  for (int l = 0; l <= 4; ++l)
    chanmix(EA, EB, F(RO + 10 + l), F(RO + 15), E9, N_EDGES, 25, 25, 9, l, (l == 0 ? 1 : 0) | (l <= 2 ? 4 : 0));
  knorm<<<N_EDGES, 256, 0, stream>>>(EB, EA, F(4), 25, 4, 0);
  { WPtrs w = {{F(11), F(12), F(13), F(14), F(15)}};
    klinmul<<<((size_t)N_EDGES * 100 + 255) / 256, 256, 0, stream>>>(EA, OF, w, F(16), N_EDGES); }
  kblockmap<<<(N_EDGES * 16 + 255) / 256, 256, 0, stream>>>(OF, CG, OB, N_EDGES);

  // --- assemble symmetric (512,512,10,10) ---
  hipMemsetAsync(d_out, 0, (size_t)out_size * sizeof(float), stream);
  kscatter_diag<<<(N_NODES * 100 + 255) / 256, 256, 0, stream>>>(DB, out);
  kscatter_off<<<N_EDGES, 128, 0, stream>>>(OB, ei, out);
  ksym<<<dim3(N_NODES, N_NODES), 128, 0, stream>>>(out);
}